// ManoLayer_7121055777372
// MI455X (gfx1250) — compile-verified
//
#include <hip/hip_runtime.h>
#include <math.h>

typedef float v2f __attribute__((ext_vector_type(2)));
typedef float v8f __attribute__((ext_vector_type(8)));

#define NB     4096
#define MESHV  778
#define NJ     16
#define NBASES 10
#define NPOSE  6
#define R3     (MESHV*3)   // 2334 rows of the blend GEMM
#define R3P    2336        // padded row count for packed D
#define KP     148         // 10 (betas) + 135 (pose_w) padded to mult of 4
#define VPST   2336        // padded VP row stride
#define MP     784         // padded mesh rows for weights

// workspace layout (floats)
#define OFF_SJ 0                          // 16*3*11 = 528
#define OFF_BK 1024                       // NB*KP
#define OFF_A  (OFF_BK + NB*KP)           // NB*256  (A^T padded: 16 cols x 16 joints)
#define OFF_GT (OFF_A  + NB*256)          // NB*48   (16 joint world translations)
#define OFF_RT (OFF_GT + NB*48)           // NB*12   (root rotation, padded)
#define OFF_DP (OFF_RT + NB*12)           // R3P*KP  (packed [shapedirs|posedirs])
#define OFF_WP (OFF_DP + R3P*KP)          // MP*16   (padded skinning weights)
#define OFF_VP (OFF_WP + MP*16)           // NB*VPST (v_posed, row-major per batch)

__device__ __forceinline__ void rodrigues3(float rx, float ry, float rz, float* R) {
  float th2 = rx*rx + ry*ry + rz*rz;
  float th  = sqrtf(th2);
  if (th < 1e-30f) {
    float a = 1.f - th2/6.f, q = 0.5f - th2/24.f;
    R[0] = 1.f + q*(rx*rx - th2); R[1] = -a*rz + q*rx*ry;       R[2] =  a*ry + q*rx*rz;
    R[3] =  a*rz + q*ry*rx;       R[4] = 1.f + q*(ry*ry - th2); R[5] = -a*rx + q*ry*rz;
    R[6] = -a*ry + q*rz*rx;       R[7] =  a*rx + q*rz*ry;       R[8] = 1.f + q*(rz*rz - th2);
  } else {
    float s = sinf(th), c = cosf(th);
    float inv = 1.f / fmaxf(th, 1e-12f);
    float nx = rx*inv, ny = ry*inv, nz = rz*inv;
    float C = 1.f - c;
    R[0] = c + C*nx*nx;     R[1] = -s*nz + C*nx*ny; R[2] =  s*ny + C*nx*nz;
    R[3] =  s*nz + C*ny*nx; R[4] = c + C*ny*ny;     R[5] = -s*nx + C*ny*nz;
    R[6] = -s*ny + C*nz*nx; R[7] =  s*nx + C*nz*ny; R[8] = c + C*nz*nz;
  }
}

// ---------- K0: fold J_regressor with shapedirs / v_template -> SJ[16][3][11] ----------
__global__ void mano_k0_fold(const float* __restrict__ Jreg,
                             const float* __restrict__ shapedirs,
                             const float* __restrict__ v_template,
                             float* __restrict__ SJ) {
  int o = blockIdx.x * blockDim.x + threadIdx.x;
  if (o >= NJ*3*11) return;
  int k = o % 11, jc = o / 11, c = jc % 3, j = jc / 3;
  float s = 0.f;
  for (int m = 0; m < MESHV; ++m) {
    float w = Jreg[j*MESHV + m];
    float v = (k < NBASES) ? shapedirs[(m*3 + c)*NBASES + k] : v_template[m*3 + c];
    s += w * v;
  }
  SJ[(j*3 + c)*11 + k] = s;
}

// ---------- KP0: pack [shapedirs|posedirs] -> Dpk[R3P][KP], weights -> Wp[MP][16] ----------
__global__ void mano_kp_pack(const float* __restrict__ sd,
                             const float* __restrict__ pd,
                             const float* __restrict__ W,
                             float* __restrict__ Dpk,
                             float* __restrict__ Wp) {
  int idx = blockIdx.x * blockDim.x + threadIdx.x;
  const int totD = R3P * KP;
  if (idx < totD) {
    int r = idx / KP, k = idx % KP;
    float v = 0.f;
    if (r < R3) {
      if (k < NBASES)   v = sd[(size_t)r*NBASES + k];
      else if (k < 145) v = pd[(size_t)r*135 + (k - NBASES)];
    }
    Dpk[idx] = v;
  } else {
    int j = idx - totD;
    if (j < MP*NJ) {
      int m = j >> 4;
      Wp[j] = (m < MESHV) ? W[j] : 0.f;
    }
  }
}

// ---------- K1: per-batch pose params, Rodrigues, kinematic chain, A^T matrices ----------
__global__ void mano_k1_prep(const float* __restrict__ root_angles,
                             const float* __restrict__ other_angles,
                             const float* __restrict__ betas,
                             const float* __restrict__ hc,   // (6,45)
                             const float* __restrict__ hm,   // (45)
                             const float* __restrict__ SJ,
                             float* __restrict__ BK, float* __restrict__ Amat,
                             float* __restrict__ Gt, float* __restrict__ Rot) {
  int b = blockIdx.x * blockDim.x + threadIdx.x;
  if (b >= NB) return;
  const int par[NJ] = {0,0,1,2,0,4,5,0,7,8,0,10,11,0,13,14};
  float be[NBASES];
  for (int k = 0; k < NBASES; ++k) be[k] = betas[b*NBASES + k];
  float oa[NPOSE];
  for (int t = 0; t < NPOSE; ++t) oa[t] = other_angles[b*NPOSE + t];
  float pb[45];
  for (int i = 0; i < 45; ++i) {
    float s = hm[i];
    for (int t = 0; t < NPOSE; ++t) s += oa[t] * hc[t*45 + i];
    pb[i] = s;
  }
  float J[NJ][3];
  for (int j = 0; j < NJ; ++j)
    for (int c = 0; c < 3; ++c) {
      const float* sj = SJ + (j*3 + c)*11;
      float s = sj[10];
      for (int k = 0; k < NBASES; ++k) s += be[k]*sj[k];
      J[j][c] = s;
    }
  float* bk = BK + (size_t)b*KP;
  for (int k = 0; k < NBASES; ++k) bk[k] = be[k];
  bk[145] = 0.f; bk[146] = 0.f; bk[147] = 0.f;

  // A stored TRANSPOSED & padded: At[col][joint], col = p*4+q (12 used, 16 padded)
  float* At = Amat + (size_t)b*256;

  float G0r[9], G0t[3];
  rodrigues3(3.14159265358979323846f, 0.f, 0.f, G0r);
  G0t[0] = J[0][0]; G0t[1] = J[0][1]; G0t[2] = J[0][2];
  for (int p = 0; p < 3; ++p) {
    float t = G0t[p] - (G0r[p*3]*J[0][0] + G0r[p*3+1]*J[0][1] + G0r[p*3+2]*J[0][2]);
    At[(p*4+0)*16 + 0] = G0r[p*3];
    At[(p*4+1)*16 + 0] = G0r[p*3+1];
    At[(p*4+2)*16 + 0] = G0r[p*3+2];
    At[(p*4+3)*16 + 0] = t;
  }
  Gt[(size_t)b*48 + 0] = G0t[0]; Gt[(size_t)b*48 + 1] = G0t[1]; Gt[(size_t)b*48 + 2] = G0t[2];

  float Gpr[9], Gpt[3];
  for (int j = 1; j < NJ; ++j) {
    int p = par[j];
    const float* Pr = (p == 0) ? G0r : Gpr;
    const float* Pt = (p == 0) ? G0t : Gpt;
    float Rj[9];
    rodrigues3(pb[(j-1)*3], pb[(j-1)*3+1], pb[(j-1)*3+2], Rj);
    for (int e = 0; e < 9; ++e) {
      float iv = (e == 0 || e == 4 || e == 8) ? 1.f : 0.f;
      bk[10 + (j-1)*9 + e] = Rj[e] - iv;            // pose_w
    }
    float d0 = J[j][0]-J[p][0], d1 = J[j][1]-J[p][1], d2 = J[j][2]-J[p][2];
    float Gr[9], Gv[3];
    for (int r = 0; r < 3; ++r) {
      for (int c = 0; c < 3; ++c)
        Gr[r*3+c] = Pr[r*3]*Rj[c] + Pr[r*3+1]*Rj[3+c] + Pr[r*3+2]*Rj[6+c];
      Gv[r] = Pr[r*3]*d0 + Pr[r*3+1]*d1 + Pr[r*3+2]*d2 + Pt[r];
    }
    for (int r = 0; r < 3; ++r) {
      float t = Gv[r] - (Gr[r*3]*J[j][0] + Gr[r*3+1]*J[j][1] + Gr[r*3+2]*J[j][2]);
      At[(r*4+0)*16 + j] = Gr[r*3];
      At[(r*4+1)*16 + j] = Gr[r*3+1];
      At[(r*4+2)*16 + j] = Gr[r*3+2];
      At[(r*4+3)*16 + j] = t;
    }
    Gt[(size_t)b*48 + j*3 + 0] = Gv[0];
    Gt[(size_t)b*48 + j*3 + 1] = Gv[1];
    Gt[(size_t)b*48 + j*3 + 2] = Gv[2];
    for (int e = 0; e < 9; ++e) Gpr[e] = Gr[e];
    Gpt[0] = Gv[0]; Gpt[1] = Gv[1]; Gpt[2] = Gv[2];
  }
  // zero padded cols 12..15 for this batch's joints
  for (int col = 12; col < 16; ++col)
    for (int j = 0; j < NJ; ++j) At[col*16 + j] = 0.f;

  float Rr[9];
  rodrigues3(root_angles[b*3], root_angles[b*3+1], root_angles[b*3+2], Rr);
  for (int e = 0; e < 9; ++e) Rot[(size_t)b*12 + e] = Rr[e];
}

// ---------- K2: v_posed = v_template + [betas|pose_w] @ Dpk^T via fp32 WMMA ----------
// one wave per 16x16 output tile; clean b64 loads + V_WMMA_F32_16X16X4_F32
__global__ void mano_k2_vposed(const float* __restrict__ BK,
                               const float* __restrict__ Dpk,
                               const float* __restrict__ v_template,
                               float* __restrict__ VP) {
  int lane = threadIdx.x;
  int hi   = lane >> 4;          // half-wave: holds K+2/K+3 of each step
  int l    = lane & 15;
  int b0 = blockIdx.y * 16;
  int r0 = blockIdx.x * 16;
  const float* ap = BK  + (size_t)(b0 + l)*KP + 2*hi;   // A row = batch
  const float* bp = Dpk + (size_t)(r0 + l)*KP + 2*hi;   // B col = (m*3+c)
  v8f acc = {};
#pragma unroll 4
  for (int kk = 0; kk < KP; kk += 4) {
    v2f a  = *(const v2f*)(ap + kk);
    v2f bb = *(const v2f*)(bp + kk);
    acc = __builtin_amdgcn_wmma_f32_16x16x4_f32(false, a, false, bb,
                                                (short)0, acc, false, false);
  }
  int bcol = r0 + l;
  if (bcol < R3) {
    float vt = v_template[bcol];
#pragma unroll
    for (int i = 0; i < 8; ++i)
      VP[(size_t)(b0 + i + 8*hi)*VPST + bcol] = acc[i] + vt;
  }
}

// ---------- K3: per-batch skinning GEMM (Wp @ A_b) via WMMA + fused apply ----------
__global__ void mano_k3_skin(const float* __restrict__ Wp,    // (784,16) padded
                             const float* __restrict__ Amat,  // (B, 16x16 A^T padded)
                             const float* __restrict__ VP,
                             const float* __restrict__ Rot,
                             float* __restrict__ outv) {
  __shared__ float ldsM[16][17];
  int lane = threadIdx.x;
  int hi   = lane >> 4;
  int l    = lane & 15;
  int b  = blockIdx.y;
  int m0 = blockIdx.x * 16;
  const float* ap = Wp + (size_t)(m0 + l)*NJ + 2*hi;        // A row = mesh vertex
  const float* bp = Amat + (size_t)b*256 + (size_t)l*16 + 2*hi; // B col l = A^T row l
  v8f acc = {};
#pragma unroll
  for (int kk = 0; kk < NJ; kk += 4) {
    v2f a  = *(const v2f*)(ap + kk);
    v2f bb = *(const v2f*)(bp + kk);
    acc = __builtin_amdgcn_wmma_f32_16x16x4_f32(false, a, false, bb,
                                                (short)0, acc, false, false);
  }
#pragma unroll
  for (int i = 0; i < 8; ++i) ldsM[i + 8*hi][l] = acc[i];   // transpose via LDS
  __syncthreads();
  if (lane < 16) {
    int m = m0 + lane;
    if (m < MESHV) {
      const float* vp = VP + (size_t)b*VPST + m*3;
      float v0 = vp[0], v1 = vp[1], v2 = vp[2];
      float x = ldsM[lane][0]*v0 + ldsM[lane][1]*v1 + ldsM[lane][2] *v2 + ldsM[lane][3];
      float y = ldsM[lane][4]*v0 + ldsM[lane][5]*v1 + ldsM[lane][6] *v2 + ldsM[lane][7];
      float z = ldsM[lane][8]*v0 + ldsM[lane][9]*v1 + ldsM[lane][10]*v2 + ldsM[lane][11];
      const float* R = Rot + (size_t)b*12;
      float* o = outv + ((size_t)b*MESHV + m)*3;
      o[0] = R[0]*x + R[1]*y + R[2]*z;
      o[1] = R[3]*x + R[4]*y + R[5]*z;
      o[2] = R[6]*x + R[7]*y + R[8]*z;
    }
  }
}

// ---------- K4: joints = rotate 16 chain translations, gather 5 tip vertices ----------
__global__ void mano_k4_joints(const float* __restrict__ Gt,
                               const float* __restrict__ Rot,
                               const float* __restrict__ verts,
                               float* __restrict__ outj) {
  int b = blockIdx.x * blockDim.x + threadIdx.x;
  if (b >= NB) return;
  const int tip_pos[5] = {4, 8, 12, 16, 20};
  const int tip_vid[5] = {333, 444, 672, 555, 745};
  const float* R = Rot + (size_t)b*12;
  int g = 0, t = 0;
  for (int pos = 0; pos < 21; ++pos) {
    float ox, oy, oz;
    if (t < 5 && pos == tip_pos[t]) {
      const float* v = verts + ((size_t)b*MESHV + tip_vid[t])*3;
      ox = v[0]; oy = v[1]; oz = v[2];
      ++t;
    } else {
      const float* gt = Gt + (size_t)b*48 + g*3;
      ox = R[0]*gt[0] + R[1]*gt[1] + R[2]*gt[2];
      oy = R[3]*gt[0] + R[4]*gt[1] + R[5]*gt[2];
      oz = R[6]*gt[0] + R[7]*gt[1] + R[8]*gt[2];
      ++g;
    }
    float* o = outj + ((size_t)b*21 + pos)*3;
    o[0] = ox; o[1] = oy; o[2] = oz;
  }
}

extern "C" void kernel_launch(void* const* d_in, const int* in_sizes, int n_in,
                              void* d_out, int out_size, void* d_ws, size_t ws_size,
                              hipStream_t stream) {
  const float* root_angles  = (const float*)d_in[0];
  const float* other_angles = (const float*)d_in[1];
  const float* betas        = (const float*)d_in[2];
  const float* v_template   = (const float*)d_in[3];
  const float* shapedirs    = (const float*)d_in[4];
  const float* posedirs     = (const float*)d_in[5];
  const float* J_regressor  = (const float*)d_in[6];
  const float* weights      = (const float*)d_in[7];
  const float* hc           = (const float*)d_in[8];
  const float* hm           = (const float*)d_in[9];

  float* ws   = (float*)d_ws;
  float* SJ   = ws + OFF_SJ;
  float* BK   = ws + OFF_BK;
  float* Amat = ws + OFF_A;
  float* Gt   = ws + OFF_GT;
  float* Rot  = ws + OFF_RT;
  float* Dpk  = ws + OFF_DP;
  float* Wp   = ws + OFF_WP;
  float* VP   = ws + OFF_VP;
  float* outv = (float*)d_out;
  float* outj = outv + (size_t)NB*MESHV*3;

  mano_k0_fold <<<dim3((NJ*3*11 + 255)/256), dim3(256), 0, stream>>>(J_regressor, shapedirs, v_template, SJ);
  mano_kp_pack <<<dim3((R3P*KP + MP*NJ + 255)/256), dim3(256), 0, stream>>>(shapedirs, posedirs, weights, Dpk, Wp);
  mano_k1_prep <<<dim3(NB/256),              dim3(256), 0, stream>>>(root_angles, other_angles, betas, hc, hm, SJ, BK, Amat, Gt, Rot);
  mano_k2_vposed<<<dim3((R3 + 15)/16, NB/16), dim3(32), 0, stream>>>(BK, Dpk, v_template, VP);
  mano_k3_skin <<<dim3((MESHV + 15)/16, NB),  dim3(32), 0, stream>>>(Wp, Amat, VP, Rot, outv);
  mano_k4_joints<<<dim3(NB/256),             dim3(256), 0, stream>>>(Gt, Rot, outv, outj);
}